// GCN_9277129359869
// MI455X (gfx1250) — compile-verified
//
#include <hip/hip_runtime.h>

typedef float v2f __attribute__((ext_vector_type(2)));
typedef float v8f __attribute__((ext_vector_type(8)));

#define N_NODES 100000
#define FEAT    256
#define NHID    64
#define NCLS    16
#define NEDGE   1600000
#define NTILES  (N_NODES / 16)   // 6250, exact

// ---------------------------------------------------------------------------
// GEMM1: support1[100000x64] = x[100000x256] @ W1[256x64], f32 WMMA 16x16x4.
// One block = one 16-row tile; 4 waves each own a 16-col slice.
// A tile (16x256 f32 = 16KB) staged in LDS once per block.
// ---------------------------------------------------------------------------
__global__ __launch_bounds__(128) void gemm1_kernel(const float* __restrict__ x,
                                                    const float* __restrict__ W1,
                                                    float* __restrict__ sup) {
    __shared__ float tileA[16 * FEAT];
    const int lane = threadIdx.x & 31;
    const int wave = threadIdx.x >> 5;
    const int rb   = blockIdx.x * 16;

    // cooperative 16KB stage: 1024 float4 loads across 128 threads
    const float4* x4 = reinterpret_cast<const float4*>(x + (size_t)rb * FEAT);
    float4* t4 = reinterpret_cast<float4*>(tileA);
    for (int i = threadIdx.x; i < 16 * FEAT / 4; i += 128) t4[i] = x4[i];
    __syncthreads();

    const int c  = wave * 16;        // column base within NHID
    const int l  = lane & 15;
    const int hi = lane >> 4;        // half-wave selects K pair (ISA 16x4 layout)
    v8f acc = {};
    for (int k = 0; k < FEAT; k += 4) {
        const int ks = k + 2 * hi;   // lanes 0-15: K=k,k+1 ; lanes 16-31: K=k+2,k+3
        v2f a = *reinterpret_cast<const v2f*>(&tileA[l * FEAT + ks]);
        v2f b;
        b.x = W1[(size_t)ks       * NHID + c + l];
        b.y = W1[(size_t)(ks + 1) * NHID + c + l];
        acc = __builtin_amdgcn_wmma_f32_16x16x4_f32(false, a, false, b,
                                                    (short)0, acc, false, false);
    }
    // D layout: VGPR r -> M=r (lanes 0-15), M=r+8 (lanes 16-31), N=lane%16
    float* out = sup + (size_t)rb * NHID + c;
    const int mb = hi * 8;
#pragma unroll
    for (int r = 0; r < 8; ++r)
        out[(size_t)(mb + r) * NHID + l] = acc[r];
}

// ---------------------------------------------------------------------------
// GEMM2: support2[100000x16] = h[100000x64] @ W2[64x16]. One wave per 16-row
// tile (N=16 exactly one WMMA tile, K=64 -> 16 chained WMMAs).
// ---------------------------------------------------------------------------
__global__ __launch_bounds__(256) void gemm2_kernel(const float* __restrict__ h,
                                                    const float* __restrict__ W2,
                                                    float* __restrict__ sup2) {
    const int lane = threadIdx.x & 31;
    const int wave = threadIdx.x >> 5;
    const int tile = blockIdx.x * 8 + wave;
    if (tile >= NTILES) return;      // wave-uniform: EXEC stays all-ones
    const int rb = tile * 16;
    const int l  = lane & 15;
    const int hi = lane >> 4;
    v8f acc = {};
    for (int k = 0; k < NHID; k += 4) {
        const int ks = k + 2 * hi;
        v2f a = *reinterpret_cast<const v2f*>(&h[(size_t)(rb + l) * NHID + ks]);
        v2f b;
        b.x = W2[ks * NCLS + l];
        b.y = W2[(ks + 1) * NCLS + l];
        acc = __builtin_amdgcn_wmma_f32_16x16x4_f32(false, a, false, b,
                                                    (short)0, acc, false, false);
    }
    float* out = sup2 + (size_t)rb * NCLS;
    const int mb = hi * 8;
#pragma unroll
    for (int r = 0; r < 8; ++r)
        out[(size_t)(mb + r) * NCLS + l] = acc[r];
}

// ---------------------------------------------------------------------------
// SPMM: acc[dst[e]][:] += w[e] * sup[src[e]][:]. F/4 threads per edge, each
// does one float4 gather (L2-resident) + 4 native f32 atomics.
// ---------------------------------------------------------------------------
template <int F>
__global__ __launch_bounds__(256) void spmm_kernel(const int* __restrict__ esrc,
                                                   const int* __restrict__ edst,
                                                   const float* __restrict__ ev,
                                                   const float* __restrict__ sup,
                                                   float* __restrict__ acc) {
    constexpr int CH = F / 4;
    const long long total  = (long long)NEDGE * CH;
    const long long stride = (long long)gridDim.x * blockDim.x;
    for (long long idx = (long long)blockIdx.x * blockDim.x + threadIdx.x;
         idx < total; idx += stride) {
        const int e    = (int)(idx / CH);
        const int part = (int)(idx % CH);
        long long pe = (idx + stride) / CH;
        if (pe >= NEDGE) pe = NEDGE - 1;
        __builtin_prefetch(&esrc[pe], 0, 0);     // global_prefetch_b8
        const int   s  = esrc[e];
        const int   d  = edst[e];
        const float wv = ev[e];
        const float4 v = reinterpret_cast<const float4*>(sup)[(size_t)s * CH + part];
        float* base = acc + (size_t)d * F + (size_t)part * 4;
        atomicAdd(base + 0, v.x * wv);
        atomicAdd(base + 1, v.y * wv);
        atomicAdd(base + 2, v.z * wv);
        atomicAdd(base + 3, v.w * wv);
    }
}

__global__ __launch_bounds__(256) void zero_kernel(float* __restrict__ p, long long n) {
    long long i = (long long)blockIdx.x * blockDim.x + threadIdx.x;
    if (i < n) p[i] = 0.0f;
}

__global__ __launch_bounds__(256) void bias_relu_kernel(float* __restrict__ h,
                                                        const float* __restrict__ b1) {
    long long i = (long long)blockIdx.x * blockDim.x + threadIdx.x;
    if (i < (long long)N_NODES * NHID) {
        const float v = h[i] + b1[i & (NHID - 1)];
        h[i] = v > 0.0f ? v : 0.0f;
    }
}

__global__ __launch_bounds__(256) void init_out_kernel(float* __restrict__ out,
                                                       const float* __restrict__ b2) {
    long long i = (long long)blockIdx.x * blockDim.x + threadIdx.x;
    if (i < (long long)N_NODES * NCLS) out[i] = b2[i & (NCLS - 1)];
}

__global__ __launch_bounds__(128) void log_softmax_kernel(float* __restrict__ out) {
    const int r = blockIdx.x * blockDim.x + threadIdx.x;
    if (r >= N_NODES) return;
    float* p = out + (size_t)r * NCLS;
    float v[NCLS];
    float m = -3.402823466e+38f;
#pragma unroll
    for (int i = 0; i < NCLS; ++i) { v[i] = p[i]; m = v[i] > m ? v[i] : m; }
    float s = 0.0f;
#pragma unroll
    for (int i = 0; i < NCLS; ++i) s += __expf(v[i] - m);
    const float lse = m + __logf(s);
#pragma unroll
    for (int i = 0; i < NCLS; ++i) p[i] = v[i] - lse;
}

// ---------------------------------------------------------------------------
extern "C" void kernel_launch(void* const* d_in, const int* in_sizes, int n_in,
                              void* d_out, int out_size, void* d_ws, size_t ws_size,
                              hipStream_t stream) {
    (void)in_sizes; (void)n_in; (void)out_size; (void)ws_size;
    const float* x   = (const float*)d_in[0];
    const int*   es  = (const int*)  d_in[1];
    const int*   ed  = (const int*)  d_in[2];
    const float* ev  = (const float*)d_in[3];
    const float* W1  = (const float*)d_in[4];
    const float* b1  = (const float*)d_in[5];
    const float* W2  = (const float*)d_in[6];
    const float* b2  = (const float*)d_in[7];
    float* out = (float*)d_out;

    // workspace: [support1: 6.4M f32][hacc/h: 6.4M f32]; support2 reuses slot 0
    float* sup1 = (float*)d_ws;
    float* hacc = sup1 + (size_t)N_NODES * NHID;
    float* sup2 = sup1;

    const long long nh = (long long)N_NODES * NHID;     // 6,400,000

    zero_kernel<<<(int)((nh + 255) / 256), 256, 0, stream>>>(hacc, nh);
    gemm1_kernel<<<NTILES, 128, 0, stream>>>(x, W1, sup1);
    {
        const long long total = (long long)NEDGE * (NHID / 4);   // 25.6M
        spmm_kernel<NHID><<<(int)((total + 255) / 256), 256, 0, stream>>>(es, ed, ev, sup1, hacc);
    }
    bias_relu_kernel<<<(int)((nh + 255) / 256), 256, 0, stream>>>(hacc, b1);
    gemm2_kernel<<<(NTILES + 7) / 8, 256, 0, stream>>>(hacc, W2, sup2);
    {
        const long long nc = (long long)N_NODES * NCLS;          // 1.6M
        init_out_kernel<<<(int)((nc + 255) / 256), 256, 0, stream>>>(out, b2);
        const long long total = (long long)NEDGE * (NCLS / 4);   // 6.4M
        spmm_kernel<NCLS><<<(int)((total + 255) / 256), 256, 0, stream>>>(es, ed, ev, sup2, out);
    }
    log_softmax_kernel<<<(N_NODES + 127) / 128, 128, 0, stream>>>(out);
}